// FasterRCNN_80255758893727
// MI455X (gfx1250) — compile-verified
//
#include <hip/hip_runtime.h>
#include <math.h>

#define RR 4096
#define MM 192
#define NN 192
#define CCH 256
#define KANCH 5
#define FIELD 16.0f

#define ROWSTRIDE (NN*CCH)          // 49152 elements between feature rows
#define PSTR 578                    // pooled row stride (even, 8B-aligned rows)

typedef float v2f __attribute__((ext_vector_type(2)));
typedef float v8f __attribute__((ext_vector_type(8)));

// -------- Kernel 1: per-ROI prolog --------------------------------------
// writes out cols 0..5 and 10..13, stores rpn boxes (cx,cy,w,h) to ws.
__global__ __launch_bounds__(256) void rpn_prolog(
    const float* __restrict__ obj_output,   // (192,192,10)
    const float* __restrict__ reg_output,   // (192,192,20)
    const float* __restrict__ anchors,      // (R,4)
    const float* __restrict__ boxes,        // (R,4)
    const int*   __restrict__ anchor_px,    // (5)
    float* __restrict__ out,                // (R,14)
    float* __restrict__ rpn)                // ws (R,4)
{
    int r = blockIdx.x * blockDim.x + threadIdx.x;
    if (r >= RR) return;
    float ax = anchors[r*4+0], ay = anchors[r*4+1], aw = anchors[r*4+3];
    int awi = (int)aw;
    int kk = 0;
    for (int i = KANCH-1; i >= 0; --i) if (anchor_px[i] == awi) kk = i;
    int px = (int)((ax - FIELD*0.5f) / FIELD);
    int py = (int)((ay - FIELD*0.5f) / FIELD);

    // obj softmax over the 2 logits of anchor kk
    const float* op = obj_output + (py*NN + px) * (2*KANCH) + kk*2;
    float o0 = op[0], o1 = op[1];
    float mx = fmaxf(o0, o1);
    float e0 = expf(o0 - mx), e1 = expf(o1 - mx);
    float inv = 1.0f / (e0 + e1);

    // reg gather + unparameterize vs anchor (ref w==h==aw)
    const float* rp = reg_output + (py*NN + px) * (4*KANCH) + kk*4;
    float t0 = rp[0], t1 = rp[1], t2 = rp[2], t3 = rp[3];
    float cx = t0 * aw + ax;
    float cy = t1 * aw + ay;
    float w  = aw * expf(t2);
    float h  = aw * expf(t3);

    // align_reg_label = parameterize(rpn_boxes, boxes)
    float bx = boxes[r*4+0], by = boxes[r*4+1], bw = boxes[r*4+2], bh = boxes[r*4+3];
    float tx = (bx + bw*0.5f - cx) / w;
    float ty = (by + bh*0.5f - cy) / h;
    float tw = logf(bw / w);
    float th = logf(bh / h);

    float* o = out + r * 14;
    o[0]  = e0 * inv;  o[1]  = e1 * inv;
    o[2]  = cx;        o[3]  = cy;       o[4]  = w;   o[5]  = h;
    o[10] = tx;        o[11] = ty;       o[12] = tw;  o[13] = th;

    rpn[r*4+0] = cx; rpn[r*4+1] = cy; rpn[r*4+2] = w; rpn[r*4+3] = h;
}

// -------- Kernel 2: ROIAlign + 2x2 maxpool + FC via f32 WMMA ------------
// One workgroup handles 16 ROIs. 256 threads = 8 wave32s.
// Channel chunks of 64; each thread processes 2 channels via b64 loads.
// Phase A fills LDS pooled[16][9][64] (row stride 578 floats, 8B-aligned),
// Phase B runs v_wmma_f32_16x16x4_f32 over disjoint K ranges per wave
// (18 k-steps each), per-wave accumulators reduced through LDS.
__global__ __launch_bounds__(256) void roialign_fc(
    const float* __restrict__ feat,   // (192,192,256)
    const float* __restrict__ Wfc,    // (2304,4)
    const float* __restrict__ bfc,    // (4)
    const float* __restrict__ rpn,    // ws (R,4)
    float* __restrict__ out)          // (R,14), writes cols 6..9
{
    __shared__ float ldsPool[16 * PSTR];  // ~36KB
    __shared__ float ldsC[8 * 256];       // 8KB partial C per wave
    __shared__ int   ldsXO[96];           // x0*C   (element offset)
    __shared__ float ldsWX[96];
    __shared__ int   ldsYO[96];           // y0*N*C (element offset)
    __shared__ float ldsWY[96];
    __shared__ float ldsBox[64];
    __shared__ float ldsReg[64];

    const int t     = threadIdx.x;
    const int wv    = t >> 5;       // wave id 0..7
    const int lane  = t & 31;
    const int rbase = blockIdx.x * 16;

    if (t < 64) ldsBox[t] = rpn[rbase*4 + t];
    if (t < 96) {
        int m = t / 6, i = t % 6;
        float cx = rpn[(rbase+m)*4+0], cy = rpn[(rbase+m)*4+1];
        float w  = rpn[(rbase+m)*4+2], h  = rpn[(rbase+m)*4+3];
        float off = ((float)i + 0.5f) / 6.0f;
        float fx = (cx - 0.5f*w + off*w) / FIELD - 0.5f;
        float fy = (cy - 0.5f*h + off*h) / FIELD - 0.5f;
        int xi = (int)fminf(fmaxf(floorf(fx), 0.0f), (float)(NN-2));
        int yi = (int)fminf(fmaxf(floorf(fy), 0.0f), (float)(MM-2));
        ldsXO[t] = xi * CCH;                 // pre-multiplied offsets: no
        ldsYO[t] = yi * ROWSTRIDE;           // per-sample multiplies later
        ldsWX[t] = fminf(fmaxf(fx - (float)xi, 0.0f), 1.0f);
        ldsWY[t] = fminf(fmaxf(fy - (float)yi, 0.0f), 1.0f);
    }

    v8f acc = {0.f,0.f,0.f,0.f,0.f,0.f,0.f,0.f};

    for (int chunk = 0; chunk < 4; ++chunk) {
        const int c0 = chunk * 64;
        __syncthreads();   // params ready / previous chunk's LDS reads done

        // ---- Phase A: bilinear sample + 2x2 max pool, 2 channels/thread --
        for (int q = 0; q < 2; ++q) {
            int p  = q * 256 + t;     // (roi, channel-pair)
            int m  = p >> 5;          // 0..15
            int cp = p & 31;          // channel pair 0..31
            int c  = c0 + cp*2;       // even -> 8B-aligned feature loads
            // hoist the 6 sample offsets/weights into registers
            int   xo[6], yo[6];
            float wxv[6], wyv[6];
#pragma unroll
            for (int i = 0; i < 6; ++i) {
                xo[i]  = ldsXO[m*6 + i] + c;   // fold channel into x offset
                yo[i]  = ldsYO[m*6 + i];
                wxv[i] = ldsWX[m*6 + i];
                wyv[i] = ldsWY[m*6 + i];
            }
#pragma unroll
            for (int ty = 0; ty < 3; ++ty) {
#pragma unroll
                for (int tx = 0; tx < 3; ++tx) {
                    v2f best = {-1.0e30f, -1.0e30f};
#pragma unroll
                    for (int sy = 0; sy < 2; ++sy) {
                        int   yoff = yo[ty*2+sy];
                        float wy   = wyv[ty*2+sy];
#pragma unroll
                        for (int sx = 0; sx < 2; ++sx) {
                            int   o  = yoff + xo[tx*2+sx];   // 32-bit math
                            float wx = wxv[tx*2+sx];
                            v2f f00 = *(const v2f*)(feat + o);
                            v2f f01 = *(const v2f*)(feat + o + CCH);
                            v2f f10 = *(const v2f*)(feat + o + ROWSTRIDE);
                            v2f f11 = *(const v2f*)(feat + o + ROWSTRIDE + CCH);
                            v2f top = f00 + wx*(f01 - f00);
                            v2f bot = f10 + wx*(f11 - f10);
                            v2f v   = top + wy*(bot - top);
                            best.x = fmaxf(best.x, v.x);
                            best.y = fmaxf(best.y, v.y);
                        }
                    }
                    *(v2f*)&ldsPool[m*PSTR + (ty*3+tx)*64 + cp*2] = best;
                }
            }
        }
        __syncthreads();

        // ---- Phase B: WMMA f32 16x16x4 over this chunk's K=576 values ----
        // wave wv owns k-steps [wv*18, wv*18+18)
        const int koff = (lane >= 16) ? 2 : 0;  // upper lane half: K+2
        const int n    = lane & 15;
        const int mrow = lane & 15;
        for (int s = 0; s < 18; ++s) {
            int kl = (wv*18 + s) * 4;
            // A operand: pooled[mrow][kl+koff .. +1]  (single b64 load)
            v2f a = *(const v2f*)&ldsPool[mrow*PSTR + kl + koff];
            // B operand: W_fc rows (zero for n>=4)
            float b0 = 0.0f, b1 = 0.0f;
            if (n < 4) {
                int klj  = kl + koff;
                int tile = klj >> 6;
                int cc   = klj & 63;
                int kg   = tile*256 + c0 + cc;       // klj+1 stays in same tile
                b0 = Wfc[kg*4 + n];
                b1 = Wfc[kg*4 + 4 + n];
            }
            v2f b; b.x = b0; b.y = b1;
            acc = __builtin_amdgcn_wmma_f32_16x16x4_f32(
                      false, a, false, b, (short)0, acc, false, false);
        }
    }

    // ---- reduce the 8 per-wave partial accumulators ----
    {
        int mbase = (lane >= 16) ? 8 : 0;
        int n = lane & 15;
#pragma unroll
        for (int v = 0; v < 8; ++v)
            ldsC[wv*256 + (v + mbase)*16 + n] = acc[v];
    }
    __syncthreads();
    {
        float s = 0.0f;
#pragma unroll
        for (int w = 0; w < 8; ++w) s += ldsC[w*256 + t];
        int m = t >> 4, n = t & 15;
        if (n < 4) ldsReg[m*4 + n] = s + bfc[n];
    }
    __syncthreads();

    // ---- align_boxes = unparameterize(align_reg, rpn_boxes) ----
    if (t < 64) {
        int m = t >> 2, j = t & 3;
        float rg = ldsReg[t];
        float bw = ldsBox[m*4+2], bh = ldsBox[m*4+3];
        float val;
        if      (j == 0) val = rg * bw + ldsBox[m*4+0];
        else if (j == 1) val = rg * bh + ldsBox[m*4+1];
        else if (j == 2) val = bw * expf(rg);
        else             val = bh * expf(rg);
        out[(rbase + m)*14 + 6 + j] = val;
    }
}

extern "C" void kernel_launch(void* const* d_in, const int* in_sizes, int n_in,
                              void* d_out, int out_size, void* d_ws, size_t ws_size,
                              hipStream_t stream) {
    (void)in_sizes; (void)n_in; (void)out_size; (void)ws_size;
    const float* features  = (const float*)d_in[0];
    const float* obj       = (const float*)d_in[1];
    const float* reg       = (const float*)d_in[2];
    const float* anchors   = (const float*)d_in[3];
    const float* boxes     = (const float*)d_in[4];
    const float* Wfc       = (const float*)d_in[5];
    const float* bfc       = (const float*)d_in[6];
    const int*   anchor_px = (const int*)d_in[7];
    float* out = (float*)d_out;
    float* rpn = (float*)d_ws;   // 4096*4 floats = 64KB scratch

    rpn_prolog<<<RR/256, 256, 0, stream>>>(obj, reg, anchors, boxes,
                                           anchor_px, out, rpn);
    roialign_fc<<<RR/16, 256, 0, stream>>>(features, Wfc, bfc, rpn, out);
}